// MMGCN_13245679141186
// MI455X (gfx1250) — compile-verified
//
#include <hip/hip_runtime.h>

#define EMB 64
#define SLOPE 0.2f

typedef __attribute__((ext_vector_type(2))) float v2f;
typedef __attribute__((ext_vector_type(8))) float v8f;

// dst[n,:] = n < n_users ? user_emb[n,:] : item_feat[n-n_users,:]
__global__ void mmgcn_concat_init(const float* __restrict__ u,
                                  const float* __restrict__ it,
                                  float* __restrict__ dst,
                                  int n_users, int total) {
    int i = blockIdx.x * blockDim.x + threadIdx.x;
    if (i >= total) return;
    int ucut = n_users * EMB;
    dst[i] = (i < ucut) ? u[i] : it[i - ucut];
}

__global__ void mmgcn_zero(float* __restrict__ p, int n) {
    int i = blockIdx.x * blockDim.x + threadIdx.x;
    if (i < n) p[i] = 0.0f;
}

// One wave32 per edge: lane handles 2 of the 64 dims.
// Y[rows[e],:] += vals[e] * X[cols[e],:]
__global__ void mmgcn_spmm(const int* __restrict__ rows,
                           const int* __restrict__ cols,
                           const float* __restrict__ vals,
                           const float* __restrict__ X,
                           float* __restrict__ Y, int n_edges) {
    int t = blockIdx.x * blockDim.x + threadIdx.x;
    int e = t >> 5;
    if (e >= n_edges) return;
    int lane = t & 31;
    int src = cols[e];
    int dst = rows[e];
    float v = vals[e];
    const v2f x = *(const v2f*)(X + (long)src * EMB + lane * 2);
    float* yd = Y + (long)dst * EMB + lane * 2;
    atomicAdd(yd + 0, v * x.x);
    atomicAdd(yd + 1, v * x.y);
}

// Y = leaky_relu(X @ W^T), X:[nrows,64], W:[64,64] row-major (out,in).
// One wave per 16-row tile; V_WMMA_F32_16X16X4_F32, K chained 16x, 4 N-tiles.
__global__ void __launch_bounds__(256)
mmgcn_gemm_lrelu(const float* __restrict__ X,
                 const float* __restrict__ W,
                 float* __restrict__ Y, int nrows) {
    __shared__ float sW[64][65];   // pad to avoid LDS bank conflicts
    int tid = threadIdx.x;
    #pragma unroll
    for (int i = tid; i < 64 * 64; i += 256)
        sW[i >> 6][i & 63] = W[i];
    __syncthreads();

    int wave = tid >> 5;
    int lane = tid & 31;
    int tile = blockIdx.x * 8 + wave;
    int row0 = tile * 16;
    if (row0 >= nrows) return;          // wave-uniform: EXEC stays all-1s for WMMA

    int half = lane >> 4;               // 0: K={0,1}, 1: K={2,3} within a k-block
    int lr   = lane & 15;               // M (for A) / N (for B)

    // Preload all 16 A fragments (one 16x4 f32 tile each): a = {X[m,k], X[m,k+1]}
    v2f a[16];
    const float* xrow = X + (long)(row0 + lr) * EMB;
    #pragma unroll
    for (int kb = 0; kb < 16; ++kb) {
        int k = kb * 4 + half * 2;
        v2f t2; t2.x = xrow[k]; t2.y = xrow[k + 1];
        a[kb] = t2;
    }

    #pragma unroll
    for (int nt = 0; nt < 4; ++nt) {
        int nb = nt * 16;
        v8f acc = {};
        #pragma unroll
        for (int kb = 0; kb < 16; ++kb) {
            int k = kb * 4 + half * 2;
            // B[k][n] = W[n][k] (y = x @ W^T)
            v2f b; b.x = sW[nb + lr][k]; b.y = sW[nb + lr][k + 1];
            acc = __builtin_amdgcn_wmma_f32_16x16x4_f32(
                false, a[kb], false, b, (short)0, acc, false, false);
        }
        // C/D layout: VGPR r -> M = half*8 + r, N = lr
        float* yb = Y + (long)(row0 + half * 8) * EMB + nb + lr;
        #pragma unroll
        for (int r = 0; r < 8; ++r) {
            float x = acc[r];
            yb[(long)r * EMB] = fmaxf(x, SLOPE * x);   // leaky_relu, slope 0.2
        }
    }
}

__global__ void mmgcn_fuse(const float* __restrict__ hv,
                           const float* __restrict__ ha,
                           const float* __restrict__ ht,
                           const float* __restrict__ uemb,
                           const float* __restrict__ iemb,
                           float* __restrict__ out,
                           int n_users, int total) {
    int i = blockIdx.x * blockDim.x + threadIdx.x;
    if (i >= total) return;
    int ucut = n_users * EMB;
    float id = (i < ucut) ? uemb[i] : iemb[i - ucut];
    out[i] = hv[i] + ha[i] + ht[i] + id;
}

extern "C" void kernel_launch(void* const* d_in, const int* in_sizes, int n_in,
                              void* d_out, int out_size, void* d_ws, size_t ws_size,
                              hipStream_t stream) {
    (void)n_in; (void)out_size; (void)ws_size;

    const int*   rows    = (const int*)d_in[0];
    const int*   cols    = (const int*)d_in[1];
    const float* vals    = (const float*)d_in[2];
    const float* user_id = (const float*)d_in[3];
    const float* item_id = (const float*)d_in[4];
    const float* u_emb[3] = {(const float*)d_in[5], (const float*)d_in[6], (const float*)d_in[7]};
    const float* i_feat[3] = {(const float*)d_in[8], (const float*)d_in[9], (const float*)d_in[10]};
    const float* Wm[3]    = {(const float*)d_in[11], (const float*)d_in[12], (const float*)d_in[13]};

    int  n_edges = in_sizes[0];
    int  n_users = in_sizes[3] / EMB;
    int  n_items = in_sizes[4] / EMB;
    int  n_nodes = n_users + n_items;
    long nelem   = (long)n_nodes * EMB;

    // Workspace: 3 persistent per-modality node-feature buffers.
    float* ws = (float*)d_ws;
    float* h[3] = { ws, ws + nelem, ws + 2 * nelem };
    // d_out doubles as the SpMM scratch; final fuse overwrites it last.
    float* tmp = (float*)d_out;

    int eb = (int)((nelem + 255) / 256);
    for (int m = 0; m < 3; ++m)
        mmgcn_concat_init<<<eb, 256, 0, stream>>>(u_emb[m], i_feat[m], h[m],
                                                  n_users, (int)nelem);

    int spmm_blocks = (int)(((long)n_edges * 32 + 255) / 256);
    int tiles       = (n_nodes + 15) / 16;
    int gemm_blocks = (tiles + 7) / 8;

    for (int l = 0; l < 2; ++l) {
        for (int m = 0; m < 3; ++m) {
            mmgcn_zero<<<eb, 256, 0, stream>>>(tmp, (int)nelem);
            mmgcn_spmm<<<spmm_blocks, 256, 0, stream>>>(rows, cols, vals,
                                                        h[m], tmp, n_edges);
            mmgcn_gemm_lrelu<<<gemm_blocks, 256, 0, stream>>>(
                tmp, Wm[m] + (long)l * EMB * EMB, h[m], n_nodes);
        }
    }

    mmgcn_fuse<<<eb, 256, 0, stream>>>(h[0], h[1], h[2], user_id, item_id,
                                       (float*)d_out, n_users, (int)nelem);
}